// VGAEModel_30562987278569
// MI455X (gfx1250) — compile-verified
//
#include <hip/hip_runtime.h>

// ---------------------------------------------------------------------------
// VGAE encoder (3x GCNConv) for MI455X / gfx1250.
// GEMMs use V_WMMA_F32_16X16X4_F32 (fp32 matrix cores, fp32 accumulate --
// the workload is scatter/atomic bound, so we keep full reference precision).
// Scatter aggregation uses global f32 atomics; per-graph xw (51MB) is L2
// resident (192MB L2), so atomics run at L2 speed, not HBM.
// ---------------------------------------------------------------------------

typedef __attribute__((ext_vector_type(2))) float v2f;
typedef __attribute__((ext_vector_type(8))) float v8f;

#define N_GRAPHS 4
#define N_NODES  50000
#define N_EDGES  800000
#define IN_CH    512
#define HID_CH   256
#define OUT_CH   128

static inline int cdiv(int a, int b) { return (a + b - 1) / b; }

// ---------------- degree / norm ----------------

__global__ void deg_init_kernel(float* __restrict__ deg, int n) {
    int i = blockIdx.x * blockDim.x + threadIdx.x;
    if (i < n) deg[i] = 1.0f;               // self-loop contributes 1
}

__global__ void deg_count_kernel(float* __restrict__ deg,
                                 const int* __restrict__ dst, int e) {
    int i = blockIdx.x * blockDim.x + threadIdx.x;
    if (i < e) atomicAdd(&deg[dst[i]], 1.0f);
}

__global__ void dinv_kernel(float* __restrict__ deg, int n) {
    int i = blockIdx.x * blockDim.x + threadIdx.x;
    if (i < n) deg[i] = rsqrtf(deg[i]);     // in place: deg -> deg^-1/2
}

__global__ void norm_kernel(float* __restrict__ norm,
                            const float* __restrict__ dinv,
                            const int* __restrict__ src,
                            const int* __restrict__ dst, int e) {
    int i = blockIdx.x * blockDim.x + threadIdx.x;
    if (i < e) norm[i] = dinv[src[i]] * dinv[dst[i]];
}

// ---------------- fp32 WMMA GEMM: C[M,N] = A[M,K] @ B[K,N] ----------------
// One wave computes a 16x64 strip of C. Per K-step of 4:
//   A (16x4 f32): lane l<16 holds A[row0+l][k0+0..1], lane 16+l holds
//   A[row0+l][k0+2..3]  -> a single float2 load per lane.
//   B (4x16 f32): mirrored layout, 4 N-tiles reuse the same A registers.
// M % 16 == 0, K % 4 == 0, N % 64 == 0 for all call sites.

__global__ void __launch_bounds__(256)
gemm_wmma_f32_kernel(const float* __restrict__ A, const float* __restrict__ B,
                     float* __restrict__ C, int M, int K, int N) {
    const int lane  = threadIdx.x & 31;
    const int wave  = blockIdx.x * (blockDim.x >> 5) + (threadIdx.x >> 5);
    const int nstr  = N >> 6;                       // 64-wide column strips
    const int row0  = (wave / nstr) << 4;
    const int col0  = (wave % nstr) << 6;
    if (row0 >= M) return;                          // uniform per wave

    const int half = lane >> 4;                     // 0: K pair 0-1, 1: K pair 2-3
    const int l    = lane & 15;

    v8f acc0 = {}, acc1 = {}, acc2 = {}, acc3 = {};

    const float* arow = A + (size_t)(row0 + l) * K + 2 * half;
    const float* bcol = B + col0 + l;

    for (int k0 = 0; k0 < K; k0 += 4) {
        v2f a = *(const v2f*)(arow + k0);
        const float* b0 = bcol + (size_t)(k0 + 2 * half) * N;
        const float* b1 = b0 + N;
        v2f bb0 = { b0[ 0], b1[ 0] };
        v2f bb1 = { b0[16], b1[16] };
        v2f bb2 = { b0[32], b1[32] };
        v2f bb3 = { b0[48], b1[48] };
        acc0 = __builtin_amdgcn_wmma_f32_16x16x4_f32(false, a, false, bb0, (short)0, acc0, false, false);
        acc1 = __builtin_amdgcn_wmma_f32_16x16x4_f32(false, a, false, bb1, (short)0, acc1, false, false);
        acc2 = __builtin_amdgcn_wmma_f32_16x16x4_f32(false, a, false, bb2, (short)0, acc2, false, false);
        acc3 = __builtin_amdgcn_wmma_f32_16x16x4_f32(false, a, false, bb3, (short)0, acc3, false, false);
    }

    // C/D layout: VGPR r -> row (row0 + r + 8*half), column col0 + 16*j + l
    float* crow = C + (size_t)(row0 + 8 * half) * N + col0 + l;
#pragma unroll
    for (int r = 0; r < 8; ++r) {
        float* cr = crow + (size_t)r * N;
        cr[ 0] = acc0[r];
        cr[16] = acc1[r];
        cr[32] = acc2[r];
        cr[48] = acc3[r];
    }
}

// ---------------- aggregation init: out[n] = bias + dinv[n]^2 * xw[n] -------

__global__ void self_init_bias_kernel(float* __restrict__ out,
                                      const float* __restrict__ xw,
                                      const float* __restrict__ dinv,
                                      const float* __restrict__ bias,
                                      int n_nodes, int f4) {
    int idx = blockIdx.x * blockDim.x + threadIdx.x;
    if (idx >= n_nodes * f4) return;
    int n = idx / f4;
    int c = idx - n * f4;
    float di = dinv[n];
    float w  = di * di;                              // self-loop norm
    float4 v = ((const float4*)xw)[(size_t)n * f4 + c];
    float4 b = ((const float4*)bias)[c];
    float4 o = { b.x + w * v.x, b.y + w * v.y, b.z + w * v.z, b.w + w * v.w };
    ((float4*)out)[(size_t)n * f4 + c] = o;
}

// ---------------- edge scatter: out[dst] += norm * xw[src] -----------------
// f4shift: log2(F/4). Each thread handles one (edge, 4-channel) pair.

__global__ void scatter_kernel(float* __restrict__ out,
                               const float* __restrict__ xw,
                               const float* __restrict__ norm,
                               const int* __restrict__ src,
                               const int* __restrict__ dst,
                               int n_edges, int f4shift) {
    int idx = blockIdx.x * blockDim.x + threadIdx.x;
    int e   = idx >> f4shift;
    if (e >= n_edges) return;
    int c   = idx & ((1 << f4shift) - 1);
    int f4  = 1 << f4shift;
    int s   = src[e];
    int d   = dst[e];
    float w = norm[e];
    float4 v = ((const float4*)xw)[(size_t)s * f4 + c];
    float* o = out + (((size_t)d * f4 + c) << 2);
    atomicAdd(o + 0, w * v.x);
    atomicAdd(o + 1, w * v.y);
    atomicAdd(o + 2, w * v.z);
    atomicAdd(o + 3, w * v.w);
}

__global__ void relu_kernel(float* __restrict__ h, int n4) {
    int i = blockIdx.x * blockDim.x + threadIdx.x;
    if (i >= n4) return;
    float4 v = ((float4*)h)[i];
    v.x = fmaxf(v.x, 0.f); v.y = fmaxf(v.y, 0.f);
    v.z = fmaxf(v.z, 0.f); v.w = fmaxf(v.w, 0.f);
    ((float4*)h)[i] = v;
}

// ---------------------------------------------------------------------------

extern "C" void kernel_launch(void* const* d_in, const int* in_sizes, int n_in,
                              void* d_out, int out_size, void* d_ws, size_t ws_size,
                              hipStream_t stream) {
    const float* x    = (const float*)d_in[0];
    const int*   eidx = (const int*)  d_in[1];
    const float* W1   = (const float*)d_in[2];
    const float* b1   = (const float*)d_in[3];
    const float* Wmu  = (const float*)d_in[4];
    const float* bmu  = (const float*)d_in[5];
    const float* Wls  = (const float*)d_in[6];
    const float* bls  = (const float*)d_in[7];
    float* out = (float*)d_out;

    // workspace layout (floats), reused across graphs: ~131 MB
    float* dinv = (float*)d_ws;                        // N_NODES
    float* nrm  = dinv + N_NODES;                      // N_EDGES
    float* xw1  = nrm  + N_EDGES;                      // N_NODES*HID_CH
    float* h    = xw1  + (size_t)N_NODES * HID_CH;     // N_NODES*HID_CH
    float* xw2  = h    + (size_t)N_NODES * HID_CH;     // N_NODES*OUT_CH

    const int B = 256;

    for (int g = 0; g < N_GRAPHS; ++g) {
        const float* xg  = x + (size_t)g * N_NODES * IN_CH;
        const int*   src = eidx + (size_t)g * 2 * N_EDGES;
        const int*   dst = src + N_EDGES;
        const float* W1g  = W1  + (size_t)g * IN_CH  * HID_CH;
        const float* Wmug = Wmu + (size_t)g * HID_CH * OUT_CH;
        const float* Wlsg = Wls + (size_t)g * HID_CH * OUT_CH;
        const float* b1g  = b1  + (size_t)g * HID_CH;
        const float* bmug = bmu + (size_t)g * OUT_CH;
        const float* blsg = bls + (size_t)g * OUT_CH;
        float* out_mu = out + (size_t)g * N_NODES * OUT_CH;
        float* out_ls = out + (size_t)(N_GRAPHS + g) * N_NODES * OUT_CH;

        // normalization (shared by all 3 convs)
        deg_init_kernel <<<cdiv(N_NODES, B), B, 0, stream>>>(dinv, N_NODES);
        deg_count_kernel<<<cdiv(N_EDGES, B), B, 0, stream>>>(dinv, dst, N_EDGES);
        dinv_kernel     <<<cdiv(N_NODES, B), B, 0, stream>>>(dinv, N_NODES);
        norm_kernel     <<<cdiv(N_EDGES, B), B, 0, stream>>>(nrm, dinv, src, dst, N_EDGES);

        // ---- conv1: h = relu(scatter(norm * (x@W1)) + b1) ----
        {
            int waves = (N_NODES / 16) * (HID_CH / 64);
            gemm_wmma_f32_kernel<<<cdiv(waves, 8), B, 0, stream>>>(
                xg, W1g, xw1, N_NODES, IN_CH, HID_CH);
        }
        self_init_bias_kernel<<<cdiv(N_NODES * (HID_CH / 4), B), B, 0, stream>>>(
            h, xw1, dinv, b1g, N_NODES, HID_CH / 4);
        scatter_kernel<<<cdiv(N_EDGES * (HID_CH / 4), B), B, 0, stream>>>(
            h, xw1, nrm, src, dst, N_EDGES, 6);        // 2^6 = 256/4
        relu_kernel<<<cdiv(N_NODES * (HID_CH / 4), B), B, 0, stream>>>(
            h, N_NODES * (HID_CH / 4));

        // ---- conv mu: out_mu = scatter(norm * (h@Wmu)) + bmu ----
        {
            int waves = (N_NODES / 16) * (OUT_CH / 64);
            gemm_wmma_f32_kernel<<<cdiv(waves, 8), B, 0, stream>>>(
                h, Wmug, xw2, N_NODES, HID_CH, OUT_CH);
        }
        self_init_bias_kernel<<<cdiv(N_NODES * (OUT_CH / 4), B), B, 0, stream>>>(
            out_mu, xw2, dinv, bmug, N_NODES, OUT_CH / 4);
        scatter_kernel<<<cdiv(N_EDGES * (OUT_CH / 4), B), B, 0, stream>>>(
            out_mu, xw2, nrm, src, dst, N_EDGES, 5);   // 2^5 = 128/4

        // ---- conv logstd: out_ls = scatter(norm * (h@Wls)) + bls ----
        {
            int waves = (N_NODES / 16) * (OUT_CH / 64);
            gemm_wmma_f32_kernel<<<cdiv(waves, 8), B, 0, stream>>>(
                h, Wlsg, xw2, N_NODES, HID_CH, OUT_CH);
        }
        self_init_bias_kernel<<<cdiv(N_NODES * (OUT_CH / 4), B), B, 0, stream>>>(
            out_ls, xw2, dinv, blsg, N_NODES, OUT_CH / 4);
        scatter_kernel<<<cdiv(N_EDGES * (OUT_CH / 4), B), B, 0, stream>>>(
            out_ls, xw2, nrm, src, dst, N_EDGES, 5);
    }
}